// SpikingGNN_46402826666012
// MI455X (gfx1250) — compile-verified
//
#include <hip/hip_runtime.h>

typedef __attribute__((ext_vector_type(2))) float v2f;
typedef __attribute__((ext_vector_type(8))) float v8f;

// ---- LIF constants (match reference) ----
#define V_REST_C   (-70.0f)
#define V_RESET_C  (-75.0f)
#define V_THRESH_C (-55.0f)
#define TAU_MEM_C  (20.0f)
#define TAU_SYN_C  (5.0f)
#define REFRAC_C   (2.0f)
#define DT_C       (1.0f)

#define B_DIM   64
#define N_DIM   4096
#define H_DIM   256
#define OUT_DIM 5

#define WMMA4(A_, B_, C_)                                                   \
  __builtin_amdgcn_wmma_f32_16x16x4_f32(false, (A_), false, (B_), (short)0, \
                                        (C_), false, false)

// ---------------------------------------------------------------------------
// fp32 WMMA GEMM: C[M,NN] = A[M,K] @ B[K,NN] (+ bias[NN] if ADD_BIAS).
// One wave32 per 16x64 C tile: 4 v8f accumulators, 8 WMMAs per K-step of 8.
// Compile-time K/NN -> immediate-offset addressing (no per-load 64-bit mul).
// A 16x4 layout: lanes 0-15 hold K={0,1}; lanes 16-31 hold K={2,3}.
// C/D layout: VGPR r, lane L -> M = r + 8*(L>=16), N = L%16.
// ---------------------------------------------------------------------------
template <int K, int NN, bool ADD_BIAS>
__global__ __launch_bounds__(32) void wmma_gemm_f32(
    const float* __restrict__ A, const float* __restrict__ Bm,
    const float* __restrict__ bias, float* __restrict__ C) {
  const int lane = threadIdx.x & 31;
  const int row  = lane & 15;
  const int hi   = lane >> 4;        // 0 -> K pair {0,1}; 1 -> K pair {2,3}
  const int n0   = blockIdx.x * 64;  // 4 N-subtiles of 16
  const int m0   = blockIdx.y * 16;

  const float* aptr = A + (size_t)(m0 + row) * K + 2 * hi;      // 8B aligned
  const float* bptr = Bm + (size_t)(2 * hi) * NN + n0 + row;

  v8f c0 = {}, c1 = {}, c2 = {}, c3 = {};

#pragma unroll 1
  for (int k0 = 0; k0 < K; k0 += 8) {
    const v2f a0 = *(const v2f*)(aptr);      // K = k0   + {0,1} (+2*hi)
    const v2f a1 = *(const v2f*)(aptr + 4);  // K = k0+4 + {0,1} (+2*hi)

    v2f b00, b01, b02, b03, b10, b11, b12, b13;
    b00.x = bptr[0 * NN +  0]; b00.y = bptr[1 * NN +  0];
    b01.x = bptr[0 * NN + 16]; b01.y = bptr[1 * NN + 16];
    b02.x = bptr[0 * NN + 32]; b02.y = bptr[1 * NN + 32];
    b03.x = bptr[0 * NN + 48]; b03.y = bptr[1 * NN + 48];
    b10.x = bptr[4 * NN +  0]; b10.y = bptr[5 * NN +  0];
    b11.x = bptr[4 * NN + 16]; b11.y = bptr[5 * NN + 16];
    b12.x = bptr[4 * NN + 32]; b12.y = bptr[5 * NN + 32];
    b13.x = bptr[4 * NN + 48]; b13.y = bptr[5 * NN + 48];

    c0 = WMMA4(a0, b00, c0);
    c1 = WMMA4(a0, b01, c1);
    c2 = WMMA4(a0, b02, c2);
    c3 = WMMA4(a0, b03, c3);
    c0 = WMMA4(a1, b10, c0);
    c1 = WMMA4(a1, b11, c1);
    c2 = WMMA4(a1, b12, c2);
    c3 = WMMA4(a1, b13, c3);

    aptr += 8;
    bptr += (size_t)8 * NN;
  }

  float bv0 = 0.f, bv1 = 0.f, bv2 = 0.f, bv3 = 0.f;
  if (ADD_BIAS) {
    bv0 = bias[n0 + row];
    bv1 = bias[n0 + 16 + row];
    bv2 = bias[n0 + 32 + row];
    bv3 = bias[n0 + 48 + row];
  }
  float* cp = C + (size_t)(m0 + 8 * hi) * NN + n0 + row;
#pragma unroll
  for (int r = 0; r < 8; ++r) {
    cp[(size_t)r * NN +  0] = c0[r] + bv0;
    cp[(size_t)r * NN + 16] = c1[r] + bv1;
    cp[(size_t)r * NN + 32] = c2[r] + bv2;
    cp[(size_t)r * NN + 48] = c3[r] + bv3;
  }
}

// ---------------------------------------------------------------------------
// LIF update (reference semantics), state held in registers per thread.
// ---------------------------------------------------------------------------
__device__ __forceinline__ float lif_step(float& v, float& isyn, float& refr,
                                          float inp) {
  const bool in_refr = refr > 0.0f;
  const float i_u = isyn + (-isyn / TAU_SYN_C + inp) * DT_C;
  const float v_u = v + (-(v - V_REST_C) + i_u) / TAU_MEM_C * DT_C;
  const bool spike = (!in_refr) && (v_u >= V_THRESH_C);
  const float v_n = in_refr ? v : (spike ? V_RESET_C : v_u);
  const float i_n = in_refr ? isyn : i_u;
  const float r_n = in_refr ? (refr - DT_C) : (spike ? REFRAC_C : refr);
  v = v_n;
  isyn = i_n;
  refr = r_n;
  return spike ? 1.0f : 0.0f;
}

// ---------------------------------------------------------------------------
// Sequential spiking loop. LIF state is shared across the batch -> T*3*B
// serial updates; only H=256 neurons are parallel. One workgroup (8 wave32),
// thread j owns neuron j of all 3 layers. W1 is staged once into LDS in full
// fp32 (256 KB of the WGP's 320 KB) to halve steady-state L2 traffic; W2
// stays L2-resident with coalesced column reads. Spike vectors via LDS.
// Dynamic LDS: [ W1 65536 | sp0 256 | sp1 256 ] floats = 264,192 B.
// ---------------------------------------------------------------------------
__global__ __launch_bounds__(H_DIM) void lif_loop(
    const float* __restrict__ x0_pre,  // [B,H]
    const float* __restrict__ W1, const float* __restrict__ b1,
    const float* __restrict__ W2, const float* __restrict__ b2,
    const float* __restrict__ Wp, const float* __restrict__ bp,
    const int* __restrict__ num_steps_p,
    float* __restrict__ hidden,   // ws [B,H]
    float* __restrict__ logits,   // out [B,OUT]
    float* __restrict__ traces) { // out [T,B,H]
  extern __shared__ float smem[];
  float* w1s = smem;                       // [H_DIM*H_DIM]
  float* sp0 = smem + H_DIM * H_DIM;       // [H_DIM]
  float* sp1 = sp0 + H_DIM;                // [H_DIM]

  const int j = threadIdx.x;  // neuron index
  const int T = *num_steps_p;

  // One-time: zero hidden accumulator, stage W1 into LDS (coalesced).
  for (int b = 0; b < B_DIM; ++b) hidden[b * H_DIM + j] = 0.0f;
#pragma unroll 4
  for (int k = 0; k < H_DIM; ++k) w1s[k * H_DIM + j] = W1[k * H_DIM + j];
  __syncthreads();

  float v0 = V_REST_C, i0 = 0.0f, r0 = 0.0f;
  float v1 = V_REST_C, i1 = 0.0f, r1 = 0.0f;
  float v2 = V_REST_C, i2 = 0.0f, r2 = 0.0f;
  const float bb1 = b1[j];
  const float bb2 = b2[j];

  for (int t = 0; t < T; ++t) {
    for (int b = 0; b < B_DIM; ++b) {
      // ---- layer 0 (input constant over t) ----
      const float x = x0_pre[b * H_DIM + j];
      const float s0 = lif_step(v0, i0, r0, x);
      sp0[j] = s0;
      __syncthreads();

      // ---- matvec sp0 @ W1 + b1 (column j, W1 in LDS) ----
      float acc1 = bb1;
#pragma unroll 8
      for (int k = 0; k < H_DIM; ++k)
        acc1 = fmaf(sp0[k], w1s[k * H_DIM + j], acc1);
      const float s1 = lif_step(v1, i1, r1, acc1);
      sp1[j] = s1;
      __syncthreads();

      // ---- matvec sp1 @ W2 + b2 (column j, W2 in L2) ----
      float acc2 = bb2;
#pragma unroll 8
      for (int k = 0; k < H_DIM; ++k)
        acc2 = fmaf(sp1[k], W2[k * H_DIM + j], acc2);
      const float s2 = lif_step(v2, i2, r2, acc2);

      traces[((size_t)t * B_DIM + b) * H_DIM + j] = s2;
      hidden[b * H_DIM + j] += s2;
      __syncthreads();  // protect sp0/sp1 reuse for next batch element
    }
  }
  __syncthreads();

  // ---- policy logits: (hidden / T) @ Wp + bp ----
  const float inv = 1.0f / (float)T;
  for (int idx = j; idx < B_DIM * OUT_DIM; idx += H_DIM) {
    const int b = idx / OUT_DIM;
    const int o = idx % OUT_DIM;
    float acc = bp[o];
    for (int k = 0; k < H_DIM; ++k)
      acc = fmaf(hidden[b * H_DIM + k] * inv, Wp[k * OUT_DIM + o], acc);
    logits[idx] = acc;
  }
}

extern "C" void kernel_launch(void* const* d_in, const int* in_sizes, int n_in,
                              void* d_out, int out_size, void* d_ws,
                              size_t ws_size, hipStream_t stream) {
  (void)in_sizes; (void)n_in; (void)out_size; (void)ws_size;

  const float* node_spikes = (const float*)d_in[0];  // [B,N]
  const float* adjacency   = (const float*)d_in[1];  // [N,N]
  const float* W0 = (const float*)d_in[2];           // [N,H]
  const float* b0 = (const float*)d_in[3];           // [H]
  const float* W1 = (const float*)d_in[4];           // [H,H]
  const float* b1 = (const float*)d_in[5];           // [H]
  const float* W2 = (const float*)d_in[6];           // [H,H]
  const float* b2 = (const float*)d_in[7];           // [H]
  const float* Wp = (const float*)d_in[8];           // [H,OUT]
  const float* bp = (const float*)d_in[9];           // [OUT]
  const int* num_steps = (const int*)d_in[10];       // scalar

  float* tmp    = (float*)d_ws;                      // [B,N]   1 MB
  float* x0_pre = tmp + (size_t)B_DIM * N_DIM;       // [B,H]   64 KB
  float* hidden = x0_pre + (size_t)B_DIM * H_DIM;    // [B,H]   64 KB

  float* logits = (float*)d_out;                     // [B,OUT]
  float* traces = logits + B_DIM * OUT_DIM;          // [T,B,H]

  // Phase 1: tmp = node_spikes @ adjacency   (WMMA f32, 16x64 tiles)
  wmma_gemm_f32<N_DIM, N_DIM, false>
      <<<dim3(N_DIM / 64, B_DIM / 16), 32, 0, stream>>>(
          node_spikes, adjacency, nullptr, tmp);

  // Phase 2: x0_pre = tmp @ W0 + b0          (WMMA f32, 16x64 tiles)
  wmma_gemm_f32<N_DIM, H_DIM, true>
      <<<dim3(H_DIM / 64, B_DIM / 16), 32, 0, stream>>>(
          tmp, W0, b0, x0_pre);

  // Phase 3: sequential LIF dynamics + traces + readout (W1 in LDS, fp32)
  const size_t lds_bytes = (size_t)(H_DIM * H_DIM + 2 * H_DIM) * sizeof(float);
  lif_loop<<<1, H_DIM, lds_bytes, stream>>>(x0_pre, W1, b1, W2, b2, Wp, bp,
                                            num_steps, hidden, logits, traces);
}